// GraphConvolution_56977036148823
// MI455X (gfx1250) — compile-verified
//
#include <hip/hip_runtime.h>

// ---- CDNA5 WMMA vector types -------------------------------------------------
typedef __attribute__((ext_vector_type(2))) float v2f;
typedef __attribute__((ext_vector_type(8))) float v8f;

#define D_FEAT 128
#define TILE_M 16

// =============================================================================
// Kernel 1: fused  h = sigmoid(x@W_gate + b) * (x @ W)
//   block = 256 threads = 8 waves; block computes a 16-row strip of h.
//   wave w computes the 16x16 output tile at columns [16w, 16w+16) using
//   V_WMMA_F32_16X16X4_F32 chained over K=128 (32 WMMAs, accumulator-carried).
//   W (128x128 fp32 = 64KB) is staged in LDS once per block.
// =============================================================================
__launch_bounds__(256, 2)
__global__ void gcn_gemm_gate_kernel(const float* __restrict__ x,
                                     const float* __restrict__ W,
                                     const float* __restrict__ Wg,
                                     const float* __restrict__ bg,
                                     float* __restrict__ h,
                                     int n_nodes) {
    __shared__ float sW[D_FEAT * D_FEAT];   // 64 KB of 320 KB WGP LDS
    __shared__ float sRed[256];
    __shared__ float sGate[TILE_M];

    const int tid  = threadIdx.x;
    const int wave = tid >> 5;      // 0..7  -> N tile
    const int lane = tid & 31;
    const int half = lane >> 4;     // 0: K={0,1} lanes, 1: K={2,3} lanes
    const int l15  = lane & 15;
    const int row0 = blockIdx.x * TILE_M;   // n_nodes % 16 == 0 for this problem

    // Prefetch this block's A rows toward L2 while we stage W (gfx1250 path).
    __builtin_prefetch(x + (size_t)(row0 + l15) * D_FEAT, 0, 1);

    // ---- stage W into LDS (float4 vectorized: 16 x b128 per thread) ----
    {
        const float4* W4  = (const float4*)W;
        float4*       sW4 = (float4*)sW;
        #pragma unroll
        for (int i = 0; i < (D_FEAT * D_FEAT / 4) / 256; ++i)
            sW4[tid + i * 256] = W4[tid + i * 256];
    }

    // ---- gate partial dot: thread t covers row (t>>4), elems [(t&15)*8, +8) ----
    {
        const int r   = tid >> 4;
        const int seg = tid & 15;
        const float* xr = x + (size_t)(row0 + r) * D_FEAT + seg * 8;
        const float* wg = Wg + seg * 8;
        float s = 0.f;
        #pragma unroll
        for (int j = 0; j < 8; ++j) s += xr[j] * wg[j];
        sRed[tid] = s;
    }
    __syncthreads();
    if (tid < TILE_M) {
        float s = bg[0];
        #pragma unroll
        for (int j = 0; j < 16; ++j) s += sRed[tid * 16 + j];
        sGate[tid] = 1.0f / (1.0f + __expf(-s));
    }
    __syncthreads();

    // ---- WMMA f32 16x16x4 over K=128, accumulator-chained ----
    const int n0 = wave * TILE_M;
    const float* arow = x + (size_t)(row0 + l15) * D_FEAT + half * 2;

    v8f acc = {};
    #pragma unroll 8
    for (int k = 0; k < D_FEAT; k += 4) {
        // A 16x4: lane l<16 -> A[l][k..k+1], lane>=16 -> A[l-16][k+2..k+3]
        v2f a = *(const v2f*)(arow + k);
        // B 4x16: VGPR v holds rows K=k+v (lanes 0-15) and K=k+v+2 (lanes 16-31)
        v2f b;
        b.x = sW[(k + half * 2 + 0) * D_FEAT + n0 + l15];
        b.y = sW[(k + half * 2 + 1) * D_FEAT + n0 + l15];
        acc = __builtin_amdgcn_wmma_f32_16x16x4_f32(
            /*neg_a=*/false, a, /*neg_b=*/false, b,
            /*c_mod=*/(short)0, acc, /*reuse_a=*/false, /*reuse_b=*/false);
    }

    // ---- epilogue: h[m][n] = gate[m] * acc; C/D VGPR v -> row m=v+half*8 ----
    float* hp = h + (size_t)row0 * D_FEAT + n0 + l15;
    #pragma unroll
    for (int v = 0; v < 8; ++v) {
        const int m = v + half * 8;
        hp[(size_t)m * D_FEAT] = acc[v] * sGate[m];
    }
}

// =============================================================================
// Kernel 2: COO scatter  out[row[e]] += vals[e] * h[col[e]]
//   one wave per edge; lane owns 4 features -> b128 gather + 4 f32 atomics.
//   h and out both live in the 192MB L2, so atomics resolve at L2 rate.
// =============================================================================
__launch_bounds__(256, 8)
__global__ void gcn_scatter_kernel(const float* __restrict__ h,
                                   const int* __restrict__ erow,
                                   const int* __restrict__ ecol,
                                   const float* __restrict__ evals,
                                   float* __restrict__ out,
                                   int n_edges) {
    const int t = blockIdx.x * blockDim.x + threadIdx.x;
    const int e = t >> 5;
    if (e >= n_edges) return;
    const int lane = t & 31;

    const int   c = ecol[e];
    const int   r = erow[e];
    const float v = evals[e];

    const float4 hv = ((const float4*)(h + (size_t)c * D_FEAT))[lane];
    float* op = out + (size_t)r * D_FEAT + lane * 4;
    atomicAdd(op + 0, v * hv.x);
    atomicAdd(op + 1, v * hv.y);
    atomicAdd(op + 2, v * hv.z);
    atomicAdd(op + 3, v * hv.w);
}

// =============================================================================
extern "C" void kernel_launch(void* const* d_in, const int* in_sizes, int n_in,
                              void* d_out, int out_size, void* d_ws, size_t ws_size,
                              hipStream_t stream) {
    const float* x     = (const float*)d_in[0];   // [N, 128]
    const int*   erow  = (const int*)  d_in[1];   // [E]
    const int*   ecol  = (const int*)  d_in[2];   // [E]
    const float* evals = (const float*)d_in[3];   // [E]
    const float* W     = (const float*)d_in[4];   // [128, 128]
    const float* Wg    = (const float*)d_in[5];   // [128, 1]
    const float* bg    = (const float*)d_in[6];   // [1]
    float*       out   = (float*)d_out;           // [N, 128]

    const int n_nodes = in_sizes[0] / D_FEAT;
    const int n_edges = in_sizes[1];

    float* h = (float*)d_ws;                      // [N, 128] scratch (51.2 MB)

    // out must start at zero for the atomic accumulation (harness poisons it).
    hipMemsetAsync(d_out, 0, (size_t)out_size * sizeof(float), stream);

    // Kernel 1: 16-row strips (100000 % 16 == 0)
    gcn_gemm_gate_kernel<<<n_nodes / TILE_M, 256, 0, stream>>>(x, W, Wg, bg, h, n_nodes);

    // Kernel 2: one wave per edge
    const long long threads = (long long)n_edges * 32;
    const int blocks = (int)((threads + 255) / 256);
    gcn_scatter_kernel<<<blocks, 256, 0, stream>>>(h, erow, ecol, evals, out, n_edges);
}